// IDWT_65790309040483
// MI455X (gfx1250) — compile-verified
//
#include <hip/hip_runtime.h>

// Inverse Haar DWT, pure HBM-streaming kernel for MI455X (gfx1250, wave32).
//
// Input : 4 subbands (N=8, C=64, H=256, W=256) f32, flat-contiguous.
// Output: (N, C, 2H=512, 2W=512) f32.
//   out[2h  ][2w] = a = (LL - HL - LH + HH)/2
//   out[2h+1][2w] = b = (LL - HL + LH - HH)/2
//   out[2h  ][2w+1] = c = (LL + HL - LH - HH)/2
//   out[2h+1][2w+1] = d = (LL + HL + LH + HH)/2
//
// Each thread processes 4 consecutive input columns of one row (one float4
// per subband, 64 B loaded) and writes 4 float4s (two adjacent 16B stores in
// each of the two corresponding output rows, 64 B stored). All accesses are
// 128-bit and wave-coalesced; non-temporal TH hints keep the 1 GB stream out
// of L2 residency.

typedef float f4 __attribute__((ext_vector_type(4)));

__global__ __launch_bounds__(256) void idwt_haar_kernel(
    const f4* __restrict__ xLL,
    const f4* __restrict__ xHL,
    const f4* __restrict__ xLH,
    const f4* __restrict__ xHH,
    f4* __restrict__ out,
    unsigned n_f4)                 // total float4s per subband
{
    const unsigned t = blockIdx.x * 256u + threadIdx.x;
    if (t >= n_f4) return;

    // Decompose flat float4 index t over (N*C, H, W/4):
    //   t = (nc << 14) | (h << 6) | w4      (W/4 = 64, H = 256)
    const unsigned w4 = t & 63u;
    const unsigned r  = t >> 6;
    const unsigned h  = r & 255u;
    const unsigned nc = r >> 8;

    const f4 ll = __builtin_nontemporal_load(xLL + t);
    const f4 hl = __builtin_nontemporal_load(xHL + t);
    const f4 lh = __builtin_nontemporal_load(xLH + t);
    const f4 hh = __builtin_nontemporal_load(xHH + t);

    // Butterfly with CSE: 8 packed adds + scale.
    const f4 p = ll + hl;   // LL+HL
    const f4 m = ll - hl;   // LL-HL
    const f4 q = lh + hh;   // LH+HH
    const f4 s = lh - hh;   // LH-HH

    const f4 a = (m - s) * 0.5f;  // even row, even col
    const f4 b = (m + s) * 0.5f;  // odd  row, even col
    const f4 c = (p - q) * 0.5f;  // even row, odd  col
    const f4 d = (p + q) * 0.5f;  // odd  row, odd  col

    // Output geometry in float4 units:
    //   per-(n,c) image: 512*512 floats = 65536 f4  -> nc << 16
    //   per output row : 512 floats     = 128   f4  -> (2h)*128 = h << 8
    //   column start   : 2*(4*w4) floats = 2*w4 f4  -> w4 << 1
    const unsigned obase = (nc << 16) | (h << 8) | (w4 << 1);

    const f4 r0lo = {a.x, c.x, a.y, c.y};
    const f4 r0hi = {a.z, c.z, a.w, c.w};
    const f4 r1lo = {b.x, d.x, b.y, d.y};
    const f4 r1hi = {b.z, d.z, b.w, d.w};

    __builtin_nontemporal_store(r0lo, out + obase);
    __builtin_nontemporal_store(r0hi, out + obase + 1);
    __builtin_nontemporal_store(r1lo, out + obase + 128);   // next output row
    __builtin_nontemporal_store(r1hi, out + obase + 129);
}

extern "C" void kernel_launch(void* const* d_in, const int* in_sizes, int n_in,
                              void* d_out, int out_size, void* d_ws, size_t ws_size,
                              hipStream_t stream) {
    (void)n_in; (void)out_size; (void)d_ws; (void)ws_size;

    const f4* xLL = (const f4*)d_in[0];
    const f4* xHL = (const f4*)d_in[1];
    const f4* xLH = (const f4*)d_in[2];
    const f4* xHH = (const f4*)d_in[3];
    f4* out = (f4*)d_out;

    const unsigned n_elems = (unsigned)in_sizes[0];   // 8*64*256*256
    const unsigned n_f4    = n_elems >> 2;            // float4s per subband
    const unsigned blocks  = (n_f4 + 255u) >> 8;

    idwt_haar_kernel<<<blocks, 256, 0, stream>>>(xLL, xHL, xLH, xHH, out, n_f4);
}